// MendGraph_20383914787275
// MI455X (gfx1250) — compile-verified
//
#include <hip/hip_runtime.h>
#include <stdint.h>

// ---------------------------------------------------------------------------
// MendGraph for MI455X (gfx1250): pure data-movement op.
//   out = [ fill_feats (N+N*P, F) f32 | fill_edges (2, E+N*P) | edge_mask ]
// ~640 MB total traffic -> ~27 us at 23.3 TB/s. Bulk concat uses the CDNA5
// async global<->LDS engine (ASYNCcnt-tracked B128 bursts, NT cache hints);
// edges/mask are a tiny VALU pass.
// ---------------------------------------------------------------------------

#define COPY_BLOCK 256
#define CHUNKS 8
#define CHUNK_BYTES (COPY_BLOCK * 16)                 // 4096 B per chunk
#define BYTES_PER_ITER (CHUNK_BYTES * CHUNKS)         // 32768 B per block-iter

__global__ __launch_bounds__(COPY_BLOCK)
void async_memcpy_kernel(const char* __restrict__ src,
                         char* __restrict__ dst,
                         unsigned long long nbytes)
{
    __shared__ __attribute__((aligned(16))) char smem[BYTES_PER_ITER];
    const unsigned tid = threadIdx.x;
    // addrspace(3) pointer value == byte offset within the workgroup LDS
    // allocation (what the async engine adds to LDS_BASE).
    const uint32_t lds0 = (uint32_t)(uintptr_t)(&smem[0]) + tid * 16u;
    const unsigned long long gstride =
        (unsigned long long)gridDim.x * (unsigned long long)BYTES_PER_ITER;

    for (unsigned long long base =
             (unsigned long long)blockIdx.x * (unsigned long long)BYTES_PER_ITER;
         base < nbytes; base += gstride)
    {
        const unsigned long long gsrc =
            (unsigned long long)(uintptr_t)src + base + (unsigned long long)tid * 16ull;
        const unsigned long long gdst =
            (unsigned long long)(uintptr_t)dst + base + (unsigned long long)tid * 16ull;
        const bool full = (base + (unsigned long long)BYTES_PER_ITER) <= nbytes;

        // ---- Phase 1: async global -> LDS B128 burst (ASYNCcnt tracked) ----
        if (full) {
#pragma unroll
            for (int c = 0; c < CHUNKS; ++c) {
                const uint32_t l = lds0 + (uint32_t)(c * CHUNK_BYTES);
                const unsigned long long ga = gsrc + (unsigned long long)(c * CHUNK_BYTES);
                asm volatile("global_load_async_to_lds_b128 %0, %1, off th:TH_LOAD_NT"
                             :: "v"(l), "v"(ga) : "memory");
            }
        } else { // uniform tail tile: per-lane predicated
#pragma unroll
            for (int c = 0; c < CHUNKS; ++c) {
                const unsigned long long o =
                    base + (unsigned long long)tid * 16ull +
                    (unsigned long long)(c * CHUNK_BYTES);
                if (o < nbytes) {
                    const uint32_t l = lds0 + (uint32_t)(c * CHUNK_BYTES);
                    const unsigned long long ga =
                        (unsigned long long)(uintptr_t)src + o;
                    asm volatile("global_load_async_to_lds_b128 %0, %1, off th:TH_LOAD_NT"
                                 :: "v"(l), "v"(ga) : "memory");
                }
            }
        }
        // LDS side of async ops may complete out of order -> drain before
        // reading the staged data back out.
        asm volatile("s_wait_asynccnt 0" ::: "memory");

        // ---- Phase 2: async LDS -> global B128 burst ----
        if (full) {
#pragma unroll
            for (int c = 0; c < CHUNKS; ++c) {
                const uint32_t l = lds0 + (uint32_t)(c * CHUNK_BYTES);
                const unsigned long long ga = gdst + (unsigned long long)(c * CHUNK_BYTES);
                asm volatile("global_store_async_from_lds_b128 %0, %1, off th:TH_STORE_NT"
                             :: "v"(ga), "v"(l) : "memory");
            }
        } else {
#pragma unroll
            for (int c = 0; c < CHUNKS; ++c) {
                const unsigned long long o =
                    base + (unsigned long long)tid * 16ull +
                    (unsigned long long)(c * CHUNK_BYTES);
                if (o < nbytes) {
                    const uint32_t l = lds0 + (uint32_t)(c * CHUNK_BYTES);
                    const unsigned long long ga =
                        (unsigned long long)(uintptr_t)dst + o;
                    asm volatile("global_store_async_from_lds_b128 %0, %1, off th:TH_STORE_NT"
                                 :: "v"(ga), "v"(l) : "memory");
                }
            }
        }
        // Drain stores before the next iteration overwrites the LDS slots.
        asm volatile("s_wait_asynccnt 0" ::: "memory");
    }
}

// fill_edges row-major [2, total] + edge_mask [total], both as f32.
__global__ __launch_bounds__(256)
void edges_mask_kernel(const int* __restrict__ edge_index,     // [2, E] flat
                       const float* __restrict__ pred_missing, // [N]
                       float* __restrict__ out_edges,          // [2, total]
                       float* __restrict__ out_mask,           // [total]
                       int N, long long E, int P, long long total)
{
    for (long long i = (long long)blockIdx.x * blockDim.x + threadIdx.x;
         i < total;
         i += (long long)gridDim.x * blockDim.x)
    {
        float s, d, m;
        if (i < E) {
            s = (float)edge_index[i];
            d = (float)edge_index[E + i];
            m = 1.0f;
        } else {
            const long long k = i - E;               // 0 .. N*P-1
            const int node = (int)(k / P);
            const int j    = (int)(k - (long long)node * P);
            s = (float)node;
            d = (float)((long long)N + k);
            int deg = (int)rintf(pred_missing[node]); // round-half-even == jnp.round
            deg = deg < 0 ? 0 : (deg > P ? P : deg);
            m = (j < deg) ? 1.0f : 0.0f;
        }
        out_edges[i]         = s;
        out_edges[total + i] = d;
        out_mask[i]          = m;
    }
}

extern "C" void kernel_launch(void* const* d_in, const int* in_sizes, int n_in,
                              void* d_out, int out_size, void* d_ws, size_t ws_size,
                              hipStream_t stream)
{
    (void)n_in; (void)out_size; (void)d_ws; (void)ws_size;

    const float* x    = (const float*)d_in[0];   // [N, F]
    const int*   ei   = (const int*)d_in[1];     // [2, E]
    const float* pred = (const float*)d_in[2];   // [N]
    const float* gen  = (const float*)d_in[3];   // [N, P*F]
    // d_in[4] = num_pred scalar; derived from shapes instead.

    const long long xCnt   = (long long)in_sizes[0];   // N*F
    const long long eiCnt  = (long long)in_sizes[1];   // 2*E
    const int       N      = in_sizes[2];
    const long long genCnt = (long long)in_sizes[3];   // N*P*F
    const long long E      = eiCnt / 2;
    const int       P      = (int)(genCnt / xCnt);

    float* out = (float*)d_out;
    float* out_feats = out;                                   // (N + N*P) * F
    const long long featCnt = xCnt + genCnt;
    float* out_edges = out + featCnt;                         // [2, total]
    const long long total = E + (long long)N * (long long)P;
    float* out_mask = out_edges + 2 * total;                  // [total]

    // ---- 1) fill_feats = concat(x, gen): two async-LDS B128 memcpys ----
    {
        const unsigned long long nb = (unsigned long long)xCnt * 4ull;
        unsigned blocks = (unsigned)((nb + BYTES_PER_ITER - 1) / BYTES_PER_ITER);
        if (blocks > 8192u) blocks = 8192u;
        if (blocks == 0u) blocks = 1u;
        async_memcpy_kernel<<<blocks, COPY_BLOCK, 0, stream>>>(
            (const char*)x, (char*)out_feats, nb);
    }
    {
        const unsigned long long nb = (unsigned long long)genCnt * 4ull;
        unsigned blocks = (unsigned)((nb + BYTES_PER_ITER - 1) / BYTES_PER_ITER);
        if (blocks > 8192u) blocks = 8192u;
        if (blocks == 0u) blocks = 1u;
        async_memcpy_kernel<<<blocks, COPY_BLOCK, 0, stream>>>(
            (const char*)gen, (char*)(out_feats + xCnt), nb);
    }

    // ---- 2) fill_edges + edge_mask ----
    {
        unsigned blocks = (unsigned)((total + 255) / 256);
        if (blocks > 8192u) blocks = 8192u;
        if (blocks == 0u) blocks = 1u;
        edges_mask_kernel<<<blocks, 256, 0, stream>>>(
            ei, pred, out_edges, out_mask, N, E, P, total);
    }
}